// GCN_Graph_44272522887307
// MI455X (gfx1250) — compile-verified
//
#include <hip/hip_runtime.h>
#include <math.h>

typedef __attribute__((ext_vector_type(2))) float v2f;
typedef __attribute__((ext_vector_type(8))) float v8f;

#define D_H 256
#define BM 128
#define BN 64
#define BK 32

// ---------------- utility kernels ----------------

__global__ void zero_f32(float* p, long long n) {
  long long i = (long long)blockIdx.x * blockDim.x + threadIdx.x;
  long long stride = (long long)gridDim.x * blockDim.x;
  for (; i < n; i += stride) p[i] = 0.0f;
}

__global__ void deg_count(const int* __restrict__ dst, float* __restrict__ cnt, int E) {
  int e = blockIdx.x * blockDim.x + threadIdx.x;
  if (e < E) atomicAdd(&cnt[dst[e]], 1.0f);
}

// cnt[i] holds raw in-degree; produce invd = 1/(1+deg), dinv = (1+deg)^-1/2
__global__ void deg_finalize(float* __restrict__ invd, float* __restrict__ dinv, int N) {
  int i = blockIdx.x * blockDim.x + threadIdx.x;
  if (i < N) {
    float d = 1.0f + invd[i];
    dinv[i] = rsqrtf(d);
    invd[i] = 1.0f / d;
  }
}

// ---------------- fp32 WMMA GEMM: C[M,N] = A[M,K] @ B[K,N] ----------------
// Requires K % 32 == 0 and N % 64 == 0 (true here: K in {128,256}, N = 256).
// Block: 256 threads = 8 waves; tile 128x64; wave -> 32x32 (2x2 WMMA tiles).

__global__ __launch_bounds__(256)
void gemm_wmma_f32(const float* __restrict__ A, const float* __restrict__ B,
                   float* __restrict__ C, int M, int K, int N) {
  __shared__ float As[BM][BK + 1];
  __shared__ float Bs[BK][BN + 1];

  const int tid  = threadIdx.x;
  const int lane = tid & 31;
  const int wid  = tid >> 5;
  const int wm   = wid >> 1;   // 0..3 -> row offset wm*32
  const int wn   = wid & 1;    // 0..1 -> col offset wn*32
  const int rowBase = blockIdx.y * BM;
  const int colBase = blockIdx.x * BN;

  v8f acc[2][2];
  #pragma unroll
  for (int a = 0; a < 2; ++a)
    #pragma unroll
    for (int b = 0; b < 2; ++b)
      #pragma unroll
      for (int v = 0; v < 8; ++v) acc[a][b][v] = 0.0f;

  for (int k0 = 0; k0 < K; k0 += BK) {
    // Stage A tile (128x32), zero-padded past M
    #pragma unroll
    for (int p = 0; p < 4; ++p) {
      int rl = p * 32 + (tid >> 3);
      int c4 = (tid & 7) * 4;
      int gr = rowBase + rl;
      float4 v = make_float4(0.f, 0.f, 0.f, 0.f);
      if (gr < M) v = *(const float4*)(A + (size_t)gr * K + k0 + c4);
      As[rl][c4 + 0] = v.x; As[rl][c4 + 1] = v.y;
      As[rl][c4 + 2] = v.z; As[rl][c4 + 3] = v.w;
    }
    // Stage B tile (32x64)
    #pragma unroll
    for (int p = 0; p < 2; ++p) {
      int rl = p * 16 + (tid >> 4);
      int c4 = (tid & 15) * 4;
      float4 v = *(const float4*)(B + (size_t)(k0 + rl) * N + colBase + c4);
      Bs[rl][c4 + 0] = v.x; Bs[rl][c4 + 1] = v.y;
      Bs[rl][c4 + 2] = v.z; Bs[rl][c4 + 3] = v.w;
    }
    __syncthreads();

    #pragma unroll
    for (int kk = 0; kk < BK; kk += 4) {
      // A 16x4 fp32 frag: lanes 0-15 hold K={kk,kk+1}, lanes 16-31 K={kk+2,kk+3}
      const int kl = kk + ((lane >> 4) << 1);
      const int ml = lane & 15;
      v2f af[2], bf[2];
      #pragma unroll
      for (int mt = 0; mt < 2; ++mt) {
        af[mt][0] = As[wm * 32 + mt * 16 + ml][kl];
        af[mt][1] = As[wm * 32 + mt * 16 + ml][kl + 1];
      }
      #pragma unroll
      for (int nt = 0; nt < 2; ++nt) {
        bf[nt][0] = Bs[kl][wn * 32 + nt * 16 + ml];
        bf[nt][1] = Bs[kl + 1][wn * 32 + nt * 16 + ml];
      }
      #pragma unroll
      for (int mt = 0; mt < 2; ++mt)
        #pragma unroll
        for (int nt = 0; nt < 2; ++nt)
          acc[mt][nt] = __builtin_amdgcn_wmma_f32_16x16x4_f32(
              false, af[mt], false, bf[nt], (short)0, acc[mt][nt], false, false);
    }
    __syncthreads();
  }

  // C/D layout: VGPR v -> row v (lanes 0-15) or v+8 (lanes 16-31), col = lane%16
  const int lrow = (lane >> 4) * 8;
  const int lcol = lane & 15;
  #pragma unroll
  for (int mt = 0; mt < 2; ++mt)
    #pragma unroll
    for (int nt = 0; nt < 2; ++nt)
      #pragma unroll
      for (int v = 0; v < 8; ++v) {
        int r = rowBase + wm * 32 + mt * 16 + lrow + v;
        int c = colBase + wn * 32 + nt * 16 + lcol;
        if (r < M) C[(size_t)r * N + c] = acc[mt][nt][v];
      }
}

// ---------------- edge aggregation: agg[dst] += H[src] * dinv[src]*dinv[dst] --------

__global__ void edge_agg(const int* __restrict__ src, const int* __restrict__ dst,
                         const float* __restrict__ dinv, const float* __restrict__ H,
                         float* __restrict__ agg, int E) {
  const int lane = threadIdx.x & 31;
  int wave = blockIdx.x * (blockDim.x >> 5) + (threadIdx.x >> 5);
  const int nwaves = gridDim.x * (blockDim.x >> 5);
  for (int e = wave; e < E; e += nwaves) {
    const int s = src[e];
    const int d = dst[e];
    const float coef = dinv[s] * dinv[d];
    const float4* hs = (const float4*)(H + (size_t)s * D_H);
    float* ad = agg + (size_t)d * D_H;
    #pragma unroll
    for (int t = 0; t < 2; ++t) {
      float4 v = hs[lane + t * 32];        // coalesced 1KB row read
      int base = (lane + t * 32) * 4;
      atomicAdd(ad + base + 0, v.x * coef);
      atomicAdd(ad + base + 1, v.y * coef);
      atomicAdd(ad + base + 2, v.z * coef);
      atomicAdd(ad + base + 3, v.w * coef);
    }
  }
}

// out = relu(agg + H * invd[node] + bias[col]); elementwise, in-place safe
__global__ void combine_relu(const float* __restrict__ agg, const float* __restrict__ H,
                             const float* __restrict__ invd, const float* __restrict__ bias,
                             float* __restrict__ out, int total) {
  int idx = blockIdx.x * blockDim.x + threadIdx.x;
  if (idx < total) {
    int i = idx >> 8;        // node
    int j = idx & (D_H - 1); // feature
    float v = agg[idx] + H[idx] * invd[i] + bias[j];
    out[idx] = fmaxf(v, 0.0f);
  }
}

__global__ void pool_add(const float* __restrict__ h, const int* __restrict__ batch,
                         float* __restrict__ g, int total) {
  int idx = blockIdx.x * blockDim.x + threadIdx.x;
  if (idx < total) {
    int i = idx >> 8;
    int j = idx & (D_H - 1);
    atomicAdd(&g[(size_t)batch[i] * D_H + j], h[idx]);
  }
}

// logits = g @ Wc + bc, then log_softmax; one wave per graph
__global__ void classify(const float* __restrict__ g, const float* __restrict__ Wc,
                         const float* __restrict__ bc, float* __restrict__ out, int G) {
  __shared__ float l[10];
  __shared__ float red[2];
  const int gi = blockIdx.x;
  const int j = threadIdx.x;
  if (j < 10) {
    float s = bc[j];
    const float* gr = g + (size_t)gi * D_H;
    for (int k = 0; k < D_H; ++k) s += gr[k] * Wc[k * 10 + j];
    l[j] = s;
  }
  __syncthreads();
  if (j == 0) {
    float m = l[0];
    for (int t = 1; t < 10; ++t) m = fmaxf(m, l[t]);
    float se = 0.0f;
    for (int t = 0; t < 10; ++t) se += expf(l[t] - m);
    red[0] = m; red[1] = logf(se);
  }
  __syncthreads();
  if (j < 10) out[gi * 10 + j] = l[j] - red[0] - red[1];
}

// ---------------- driver ----------------

extern "C" void kernel_launch(void* const* d_in, const int* in_sizes, int n_in,
                              void* d_out, int out_size, void* d_ws, size_t ws_size,
                              hipStream_t stream) {
  const float* x     = (const float*)d_in[0];
  // d_in[1] = edge_attr (unused by the module)
  const int*   eidx  = (const int*)d_in[2];
  const int*   batch = (const int*)d_in[3];
  const float* W1 = (const float*)d_in[4];  const float* b1 = (const float*)d_in[5];
  const float* W2 = (const float*)d_in[6];  const float* b2 = (const float*)d_in[7];
  const float* W3 = (const float*)d_in[8];  const float* b3 = (const float*)d_in[9];
  const float* Wc = (const float*)d_in[10]; const float* bc = (const float*)d_in[11];

  const int N = in_sizes[0] / 128;   // nodes
  const int E = in_sizes[2] / 2;     // edges
  const int G = out_size / 10;       // graphs

  const int* src = eidx;
  const int* dst = eidx + E;

  float* ws = (float*)d_ws;
  const size_t nd = (size_t)N * D_H;
  float* bufA  = ws;               // N*256
  float* bufB  = ws + nd;          // N*256
  float* invd  = ws + 2 * nd;      // N  (first: raw degree counts; then 1/deg)
  float* dinv  = invd + N;         // N  (deg^-1/2)
  float* gpool = dinv + N;         // G*256

  const int ethreads = (E + 255) / 256;
  const int nthreads = (N + 255) / 256;
  const int ndblocks = (int)((nd + 255) / 256);
  dim3 ggrid(D_H / BN, (N + BM - 1) / BM);

  // degree tables (shared by all layers)
  zero_f32<<<256, 256, 0, stream>>>(invd, N);
  deg_count<<<ethreads, 256, 0, stream>>>(dst, invd, E);
  deg_finalize<<<nthreads, 256, 0, stream>>>(invd, dinv, N);

  // ---- layer 1: H1 = x @ W1 ; agg ; h1 = relu(...)  (h1 -> bufA)
  gemm_wmma_f32<<<ggrid, 256, 0, stream>>>(x, W1, bufA, N, 128, D_H);
  zero_f32<<<4096, 256, 0, stream>>>(bufB, (long long)nd);
  edge_agg<<<8192, 256, 0, stream>>>(src, dst, dinv, bufA, bufB, E);
  combine_relu<<<ndblocks, 256, 0, stream>>>(bufB, bufA, invd, b1, bufA, (int)nd);

  // ---- layer 2 (h2 -> bufB)
  gemm_wmma_f32<<<ggrid, 256, 0, stream>>>(bufA, W2, bufB, N, 256, D_H);
  zero_f32<<<4096, 256, 0, stream>>>(bufA, (long long)nd);
  edge_agg<<<8192, 256, 0, stream>>>(src, dst, dinv, bufB, bufA, E);
  combine_relu<<<ndblocks, 256, 0, stream>>>(bufA, bufB, invd, b2, bufB, (int)nd);

  // ---- layer 3 (h3 -> bufA)
  gemm_wmma_f32<<<ggrid, 256, 0, stream>>>(bufB, W3, bufA, N, 256, D_H);
  zero_f32<<<4096, 256, 0, stream>>>(bufB, (long long)nd);
  edge_agg<<<8192, 256, 0, stream>>>(src, dst, dinv, bufA, bufB, E);
  combine_relu<<<ndblocks, 256, 0, stream>>>(bufB, bufA, invd, b3, bufA, (int)nd);

  // ---- pool + classifier + log_softmax
  zero_f32<<<64, 256, 0, stream>>>(gpool, (long long)G * D_H);
  pool_add<<<ndblocks, 256, 0, stream>>>(bufA, batch, gpool, (int)nd);
  classify<<<G, 32, 0, stream>>>(gpool, Wc, bc, (float*)d_out, G);
}